// Circuit_19275813225041
// MI455X (gfx1250) — compile-verified
//
#include <hip/hip_runtime.h>

// ---------------------------------------------------------------------------
// 24-qubit statevector circuit on MI455X (gfx1250).
// Two LDS-resident sweeps. Gates fused into 12 passes of 8x8-COMPLEX
// operators (2 ladder gates + absorbed 1q gates) = 16x16 real matmul
// = 4 chained V_WMMA_F32_16X16X4_F32 at 100% row utilization.
// ---------------------------------------------------------------------------

#define NQ      24
#define DIM     (1 << NQ)
#define TB      15                    // tile = 2^15 amplitudes
#define TILE    (1 << TB)
#define THREADS 512                   // 16 waves (wave32)
#define NWAVES  (THREADS / 32)
#define LDSPLANE (TILE + (TILE >> 6)) // +1 word pad per 64 -> 33280 words
#define LIMOFF   (LDSPLANE + 16)      // +16 words: im plane bank-shifted by 16
#define LDSBYTES ((LDSPLANE + LIMOFF) * 4)     // 266,304 B <= 320 KB

typedef __attribute__((ext_vector_type(2))) float v2f;
typedef __attribute__((ext_vector_type(8))) float v8f;

typedef __attribute__((address_space(1))) int ga_int;   // global
typedef __attribute__((address_space(3))) int ls_int;   // LDS

#if defined(__HIP_DEVICE_COMPILE__)
#if !__has_builtin(__builtin_amdgcn_wmma_f32_16x16x4_f32)
#error "__builtin_amdgcn_wmma_f32_16x16x4_f32 not available on this toolchain"
#endif
#if __has_builtin(__builtin_amdgcn_global_load_async_to_lds_b32)
#define USE_ASYNC_LDS 1
#endif
#if __has_builtin(__builtin_amdgcn_s_wait_asynccnt)
#define HAVE_WAIT_ASYNC_BUILTIN 1
#endif
#endif

__device__ __forceinline__ int pad(int i) { return i + (i >> 6); }

// Fused pass table: {local p (lo bit), qa (gates2 idx on (hi,mid); -1=I),
//                    qb (gates2 idx on (mid,lo)), qh, qm, ql (gates1; -1=I)}
// U = (I (x) G_qb) * (G_qa (x) I) * (k_qh (x) k_qm (x) k_ql)   on bits (p+2,p+1,p)
// Passes 0..4: sweep 0 (local bits 14..4 <-> global 23..13, local 3..0 <-> global 3..0)
// Passes 5..11: sweep 1 (local == global bits 14..0)
constexpr int FPASS[12][6] = {
    {12,  0,  1,  0,  1,  2},
    {10,  2,  3, -1,  3,  4},
    { 8,  4,  5, -1,  5,  6},
    { 6,  6,  7, -1,  7,  8},
    { 4,  8,  9, -1,  9, 10},
    {11, 10, 11, -1, 11, 12},
    { 9, 12, 13, -1, 13, 14},
    { 7, 14, 15, -1, 15, 16},
    { 5, 16, 17, -1, 17, 18},
    { 3, 18, 19, -1, 19, 20},
    { 1, 20, 21, -1, 21, 22},
    { 0, -1, 22, -1, -1, 23},
};

// complex entry of a 2x2 gate (gates1[q]); identity if q < 0
__device__ __forceinline__ void e1(const float* g1, int q, int r, int c,
                                   float& er, float& ei) {
    if (q < 0) { er = (r == c) ? 1.f : 0.f; ei = 0.f; }
    else       { er = g1[q*8 + r*2 + c];    ei = g1[q*8 + 4 + r*2 + c]; }
}
// complex entry of a 4x4 gate (gates2[q]); identity if q < 0
__device__ __forceinline__ void e2(const float* g2, int q, int r, int c,
                                   float& er, float& ei) {
    if (q < 0) { er = (r == c) ? 1.f : 0.f; ei = 0.f; }
    else       { er = g2[q*32 + r*4 + c];   ei = g2[q*32 + 16 + r*4 + c]; }
}

template <int SWEEP>
__global__ __launch_bounds__(THREADS)
void qsim_sweep(const float* __restrict__ in, const float* __restrict__ g1,
                const float* __restrict__ g2, float* __restrict__ out)
{
    extern __shared__ float smem[];
    float* lre = smem;
    float* lim = smem + LIMOFF;

    const int tid  = threadIdx.x;
    const int lane = tid & 31;
    const int wave = tid >> 5;
    const int blk  = blockIdx.x;

    // ---------------- fill LDS tile ----------------
#if defined(USE_ASYNC_LDS)
    for (int it = 0; it < TILE / THREADS; ++it) {
        int v = it * THREADS + tid;
        int g = SWEEP ? ((blk << TB) | v)
                      : (((v >> 4) << 13) | (blk << 4) | (v & 15));
        __builtin_amdgcn_global_load_async_to_lds_b32(
            (ga_int*)(in + g), (ls_int*)&lre[pad(v)], 0, 0);
        __builtin_amdgcn_global_load_async_to_lds_b32(
            (ga_int*)(in + DIM + g), (ls_int*)&lim[pad(v)], 0, 0);
    }
#if defined(HAVE_WAIT_ASYNC_BUILTIN)
    __builtin_amdgcn_s_wait_asynccnt(0);
#else
    asm volatile("s_wait_asynccnt 0" ::: "memory");
#endif
#else
    for (int it = 0; it < TILE / (THREADS * 4); ++it) {
        int v = (it * THREADS + tid) << 2;
        int g = SWEEP ? ((blk << TB) | v)
                      : (((v >> 4) << 13) | (blk << 4) | (v & 15));
        float4 r = *(const float4*)(in + g);
        float4 m = *(const float4*)(in + DIM + g);
        lre[pad(v+0)] = r.x; lre[pad(v+1)] = r.y; lre[pad(v+2)] = r.z; lre[pad(v+3)] = r.w;
        lim[pad(v+0)] = m.x; lim[pad(v+1)] = m.y; lim[pad(v+2)] = m.z; lim[pad(v+3)] = m.w;
    }
#endif
    __syncthreads();

    // ---------------- fused gate passes ----------------
    // A layout (16x4 f32): lanes 0-15 -> M=lane, K={0,1}; lanes 16-31 -> M=lane-16, K={2,3}
    const int mreal = lane & 15;          // A row (16x16 real matrix row) / D column N
    const int mc    = mreal & 7;          // complex output amplitude index
    const int outIm = mreal >> 3;         // 0: re-output rows, 1: im-output rows
    const int kb    = (lane >> 4) << 1;   // K base within each 4-chunk: 0 or 2

#pragma unroll
    for (int ip = (SWEEP ? 5 : 0); ip < (SWEEP ? 12 : 5); ++ip) {
        const int p  = FPASS[ip][0];
        const int qa = FPASS[ip][1], qb = FPASS[ip][2];
        const int qh = FPASS[ip][3], qm = FPASS[ip][4], ql = FPASS[ip][5];

        // ---- per-lane entries of U (8x8 complex): U[mc, j], j in {kb,kb+1,kb+4,kb+5} ----
        // P[t] = sum_{s1} B[(m1,m0),(s1,t0)] * A[(m2,s1),(t2,t1)]
        const int m0 = mc & 1, m1 = (mc >> 1) & 1, m2 = mc >> 2;
        float Pr[8], Pi[8];
#pragma unroll
        for (int t = 0; t < 8; ++t) {
            float pr = 0.f, pi = 0.f;
#pragma unroll
            for (int s1 = 0; s1 < 2; ++s1) {
                float br, bi, ar, ai;
                e2(g2, qb, m1*2 + m0, s1*2 + (t & 1), br, bi);
                e2(g2, qa, m2*2 + s1, t >> 1, ar, ai);
                pr += br*ar - bi*ai;  pi += br*ai + bi*ar;
            }
            Pr[t] = pr; Pi[t] = pi;
        }
        // U[mc,k] = sum_t P[t] * S[t,k],  S[t,k] = kh[t2,k2]*km[t1,k1]*kl[t0,k0]
        float Ur[4], Ui[4];
#pragma unroll
        for (int jj = 0; jj < 4; ++jj) {
            const int k = ((jj >> 1) << 2) | kb | (jj & 1);
            float ur = 0.f, ui = 0.f;
#pragma unroll
            for (int t = 0; t < 8; ++t) {
                float hr, hi, mr, mi, lr, li;
                e1(g1, qh, t >> 2, k >> 2, hr, hi);
                e1(g1, qm, (t >> 1) & 1, (k >> 1) & 1, mr, mi);
                e1(g1, ql, t & 1, k & 1, lr, li);
                const float hmr = hr*mr - hi*mi, hmi = hr*mi + hi*mr;
                const float sr  = hmr*lr - hmi*li, si = hmr*li + hmi*lr;
                ur += Pr[t]*sr - Pi[t]*si;
                ui += Pr[t]*si + Pi[t]*sr;
            }
            Ur[jj] = ur; Ui[jj] = ui;
        }
        // A-matrix chunks of R16 = [[Ur,-Ui],[Ui,Ur]] (K chunks: re0-3, re4-7, im0-3, im4-7)
        v2f A0, A1, A2, A3;
        if (outIm == 0) {
            A0[0] =  Ur[0]; A0[1] =  Ur[1];  A1[0] =  Ur[2]; A1[1] =  Ur[3];
            A2[0] = -Ui[0]; A2[1] = -Ui[1];  A3[0] = -Ui[2]; A3[1] = -Ui[3];
        } else {
            A0[0] =  Ui[0]; A0[1] =  Ui[1];  A1[0] =  Ui[2]; A1[1] =  Ui[3];
            A2[0] =  Ur[0]; A2[1] =  Ur[1];  A3[0] =  Ur[2]; A3[1] =  Ur[3];
        }

        // ---- batch loop: 16 groups of 8 amplitudes per WMMA-quad ----
        const int sp    = 1 << p;
        const int oA    = (kb >> 1) << (p + 1);  // off(kb); +sp, +4sp, +5sp for the rest
        const int pmask = sp - 1;
        float* pl = outIm ? lim : lre;           // this lane's output plane

#pragma unroll 2
        for (int batch = wave; batch < (TILE >> 7); batch += NWAVES) {
            const int g    = (batch << 4) + mreal;               // group id
            const int base = ((g >> p) << (p + 3)) | (g & pmask);
            const int b0 = pad(base + oA);
            const int b1 = pad(base + oA + sp);
            const int b4 = pad(base + oA + 4*sp);
            const int b5 = pad(base + oA + 5*sp);
            v2f B0; B0[0] = lre[b0]; B0[1] = lre[b1];
            v2f B1; B1[0] = lre[b4]; B1[1] = lre[b5];
            v2f B2; B2[0] = lim[b0]; B2[1] = lim[b1];
            v2f B3; B3[0] = lim[b4]; B3[1] = lim[b5];
            v8f d = {0.f,0.f,0.f,0.f,0.f,0.f,0.f,0.f};
#if defined(__HIP_DEVICE_COMPILE__)
            d = __builtin_amdgcn_wmma_f32_16x16x4_f32(false, A0, false, B0, (short)0, d, false, false);
            d = __builtin_amdgcn_wmma_f32_16x16x4_f32(false, A1, false, B1, (short)0, d, false, false);
            d = __builtin_amdgcn_wmma_f32_16x16x4_f32(false, A2, false, B2, (short)0, d, false, false);
            d = __builtin_amdgcn_wmma_f32_16x16x4_f32(false, A3, false, B3, (short)0, d, false, false);
#endif
            // D: lane<16 holds re-out comps 0..7 of group mreal; lane>=16 holds im-out comps
#pragma unroll
            for (int r = 0; r < 8; ++r) {
                const int offr = ((r >> 2) << (p + 2)) | (((r >> 1) & 1) << (p + 1)) | ((r & 1) << p);
                pl[pad(base + offr)] = d[r];
            }
        }
        __syncthreads();
    }

    // ---------------- drain LDS tile ----------------
    for (int it = 0; it < TILE / (THREADS * 4); ++it) {
        int v = (it * THREADS + tid) << 2;
        int g = SWEEP ? ((blk << TB) | v)
                      : (((v >> 4) << 13) | (blk << 4) | (v & 15));
        float4 r, m;
        r.x = lre[pad(v)];  r.y = lre[pad(v+1)]; r.z = lre[pad(v+2)]; r.w = lre[pad(v+3)];
        m.x = lim[pad(v)];  m.y = lim[pad(v+1)]; m.z = lim[pad(v+2)]; m.w = lim[pad(v+3)];
        *(float4*)(out + g)       = r;
        *(float4*)(out + DIM + g) = m;
    }
}

extern "C" void kernel_launch(void* const* d_in, const int* in_sizes, int n_in,
                              void* d_out, int out_size, void* d_ws, size_t ws_size,
                              hipStream_t stream) {
    (void)in_sizes; (void)n_in; (void)d_ws; (void)ws_size; (void)out_size;
    const float* state = (const float*)d_in[0];   // (2, 2^24) f32
    const float* g1    = (const float*)d_in[1];   // (24, 2, 2, 2) f32
    const float* g2    = (const float*)d_in[2];   // (23, 2, 4, 4) f32
    float* out = (float*)d_out;

    (void)hipFuncSetAttribute((const void*)qsim_sweep<0>,
                              hipFuncAttributeMaxDynamicSharedMemorySize, LDSBYTES);
    (void)hipFuncSetAttribute((const void*)qsim_sweep<1>,
                              hipFuncAttributeMaxDynamicSharedMemorySize, LDSBYTES);

    // Sweep 0: bits {23..13}|{3..0}, in -> out.  Sweep 1: bits {14..0}, out in-place.
    qsim_sweep<0><<<DIM / TILE, THREADS, LDSBYTES, stream>>>(state, g1, g2, out);
    qsim_sweep<1><<<DIM / TILE, THREADS, LDSBYTES, stream>>>(out,   g1, g2, out);
}